// DynamicPillarFeatureNet_36919538876792
// MI455X (gfx1250) — compile-verified
//
#include <hip/hip_runtime.h>
#include <hip/hip_bf16.h>

// ---------------- Geometry constants (from reference) ----------------
#define NYV 400
#define NXV 352
#define BATCH 4
#define NUM_VOX (BATCH * NYV * NXV)      // 563200
#define C_OUT 64
#define BN_EPS 1e-3f
#define VXs 0.2f
#define VYs 0.2f
#define VZs 4.0f
#define X_OFF 0.1f        // VX/2 + pc_range[0]
#define Y_OFF (-39.9f)    // VY/2 + pc_range[1]
#define Z_OFF (-1.0f)     // VZ/2 + pc_range[2]

typedef __attribute__((ext_vector_type(2))) float v2f;
typedef __attribute__((ext_vector_type(8))) float v8f;

// ---------------------------------------------------------------------------
// Kernel 1: scatter-add raw features + counts into pillar accumulators.
// 2M points * (4 feature atomics + 1 count atomic). Targets stay L2-resident
// (563200*5 floats = 11 MB << 192 MB L2), so atomic RMW happens at L2 rate.
// ---------------------------------------------------------------------------
__global__ void pillar_scatter_add(const float4* __restrict__ feat,
                                   const int4* __restrict__ coors,
                                   float* __restrict__ sums,   // [NUM_VOX*4]
                                   float* __restrict__ cnt,    // [NUM_VOX]
                                   int n) {
    int stride = gridDim.x * blockDim.x;
    for (int i = blockIdx.x * blockDim.x + threadIdx.x; i < n; i += stride) {
        float4 f = feat[i];
        int4 c = coors[i];                       // (b, z, y, x)
        int vox = c.x * (NYV * NXV) + c.z * NXV + c.w;
        float* s = sums + (size_t)vox * 4;
        atomicAdd(s + 0, f.x);
        atomicAdd(s + 1, f.y);
        atomicAdd(s + 2, f.z);
        atomicAdd(s + 3, f.w);
        atomicAdd(cnt + vox, 1.0f);
    }
}

// ---------------------------------------------------------------------------
// Kernel 2: fused gather-mean -> augment(10ch) -> GEMM(10x64, f32 WMMA)
//           -> BN(eval)+ReLU -> u32 atomic-max scatter into dense canvas.
//
// One wave32 = 16 points. f32 A-matrix (16x4) layout: lanes 0-15 hold K={0,1},
// lanes 16-31 hold K={2,3}; both half-waves compute the same point's 10-vector
// (scalar regs, explicit hi/lo selects -> no dynamic private-array indexing)
// and slice out their K pair. K=10 padded to 12 -> 3 WMMA chunks per 16x16
// output tile, 4 tiles across the 64 output channels = 12 v_wmma per group.
// B fragments (W, 2.5KB) + BN scale/bias hoisted out of a grid-stride loop.
// ReLU => h >= 0 => float bits monotone as u32 => atomicMax(u32) against a
// zero-filled canvas == segment_max + empty-pillar zeroing.
// ---------------------------------------------------------------------------
__global__ void pfn_wmma_fused(const float4* __restrict__ feat,
                               const int4* __restrict__ coors,
                               const float* __restrict__ W,       // [10,64] row-major
                               const float* __restrict__ gamma,
                               const float* __restrict__ beta,
                               const float* __restrict__ bn_mean,
                               const float* __restrict__ bn_var,
                               const float* __restrict__ sums,
                               const float* __restrict__ cnt,
                               float* __restrict__ out,           // [NUM_VOX,64]
                               int n) {
    const int lane = threadIdx.x & 31;
    const int wave = threadIdx.x >> 5;
    const int gwave = blockIdx.x * (blockDim.x >> 5) + wave;
    const int nwaves = gridDim.x * (blockDim.x >> 5);
    const bool hi = lane >= 16;
    const int col = lane & 15;
    const int kb = hi ? 2 : 0;

    // ---- hoisted B fragments: Bf[tile][chunk] covers K=4*chunk..+3, N=16*tile..+15
    v2f Bf[4][3];
#pragma unroll
    for (int nt = 0; nt < 4; ++nt) {
#pragma unroll
        for (int c = 0; c < 3; ++c) {
            int k0 = 4 * c + kb;
            float bx = (k0 < 10)     ? W[k0 * C_OUT + nt * 16 + col]       : 0.0f;
            float by = (k0 + 1 < 10) ? W[(k0 + 1) * C_OUT + nt * 16 + col] : 0.0f;
            Bf[nt][c] = (v2f){bx, by};
        }
    }
    // ---- hoisted BN affine per output column
    float scl[4], bia[4];
#pragma unroll
    for (int nt = 0; nt < 4; ++nt) {
        int ch = nt * 16 + col;
        float s = gamma[ch] * rsqrtf(bn_var[ch] + BN_EPS);
        scl[nt] = s;
        bia[nt] = beta[ch] - bn_mean[ch] * s;
    }

    const int ngroups = (n + 15) >> 4;
    for (int g = gwave; g < ngroups; g += nwaves) {
        const int p = g * 16 + col;           // both half-waves mirror the same point
        const bool fullGroup = (g * 16 + 16) <= n;
        const bool pv = p < n;

        float4 f = pv ? feat[p] : (float4){0.f, 0.f, 0.f, 0.f};
        int4 cr = pv ? coors[p] : (int4){0, 0, 0, 0};
        int vox = cr.x * (NYV * NXV) + cr.z * NXV + cr.w;

        // cnt is a small exact integer in f32: hardware v_rcp_f32 (<=1 ulp,
        // exact for pow2) replaces the ~10-instr IEEE division expansion.
        float inv = __builtin_amdgcn_rcpf(fmaxf(cnt[vox], 1.0f));
        const float* s4 = sums + (size_t)vox * 4;
        // 10-channel augmented feature, all in named scalars (no dyn indexing)
        float a0 = f.x, a1 = f.y, a2 = f.z, a3 = f.w;               // raw (4)
        float a4 = f.x - s4[0] * inv;                               // cluster (3)
        float a5 = f.y - s4[1] * inv;
        float a6 = f.z - s4[2] * inv;
        float a7 = f.x - ((float)cr.w * VXs + X_OFF);               // center (3)
        float a8 = f.y - ((float)cr.z * VYs + Y_OFF);
        float a9 = f.z - ((float)cr.y * VZs + Z_OFF);

        // A fragments: lo half-wave takes K pair {4c,4c+1}, hi takes {4c+2,4c+3}
        v2f A0 = hi ? (v2f){a2, a3} : (v2f){a0, a1};
        v2f A1 = hi ? (v2f){a6, a7} : (v2f){a4, a5};
        v2f A2 = hi ? (v2f){0.f, 0.f} : (v2f){a8, a9};   // K=10,11 are zero pad

        // voxel id of the row each C/D register r maps to (M = r + 8*hi)
        int voxm[8];
#pragma unroll
        for (int r = 0; r < 8; ++r)
            voxm[r] = __shfl(vox, r + (hi ? 8 : 0), 32);

        // 12 WMMAs: 4 output tiles x 3 K-chunks, accumulator chained
        v8f acc[4];
#pragma unroll
        for (int nt = 0; nt < 4; ++nt) {
            v8f t = (v8f){0.f, 0.f, 0.f, 0.f, 0.f, 0.f, 0.f, 0.f};
            t = __builtin_amdgcn_wmma_f32_16x16x4_f32(false, A0, false, Bf[nt][0],
                                                      (short)0, t, false, false);
            t = __builtin_amdgcn_wmma_f32_16x16x4_f32(false, A1, false, Bf[nt][1],
                                                      (short)0, t, false, false);
            t = __builtin_amdgcn_wmma_f32_16x16x4_f32(false, A2, false, Bf[nt][2],
                                                      (short)0, t, false, false);
            acc[nt] = t;
        }

        // Epilogue: BN + ReLU + u32 atomic-max scatter. Row-major so the
        // 64-bit canvas base is computed once per row; unguarded fast path
        // for full groups (N % 16 == 0 => always taken in practice).
        if (fullGroup) {
#pragma unroll
            for (int r = 0; r < 8; ++r) {
                unsigned int* base =
                    (unsigned int*)(out + (size_t)voxm[r] * C_OUT + col);
#pragma unroll
                for (int nt = 0; nt < 4; ++nt) {
                    float h = fmaxf(fmaf(acc[nt][r], scl[nt], bia[nt]), 0.0f);
                    atomicMax(base + nt * 16, __float_as_uint(h));
                }
            }
        } else {
#pragma unroll
            for (int r = 0; r < 8; ++r) {
                if (g * 16 + r + (hi ? 8 : 0) < n) {
                    unsigned int* base =
                        (unsigned int*)(out + (size_t)voxm[r] * C_OUT + col);
#pragma unroll
                    for (int nt = 0; nt < 4; ++nt) {
                        float h = fmaxf(fmaf(acc[nt][r], scl[nt], bia[nt]), 0.0f);
                        atomicMax(base + nt * 16, __float_as_uint(h));
                    }
                }
            }
        }
    }
}

extern "C" void kernel_launch(void* const* d_in, const int* in_sizes, int n_in,
                              void* d_out, int out_size, void* d_ws, size_t ws_size,
                              hipStream_t stream) {
    const float4* feat  = (const float4*)d_in[0];
    const int4*   coors = (const int4*)d_in[1];
    const float*  W     = (const float*)d_in[2];
    const float*  gamma = (const float*)d_in[3];
    const float*  beta  = (const float*)d_in[4];
    const float*  bmean = (const float*)d_in[5];
    const float*  bvar  = (const float*)d_in[6];
    float* out = (float*)d_out;

    const int n = in_sizes[0] / 4;              // 2,000,000 points

    float* sums = (float*)d_ws;                 // [NUM_VOX*4]
    float* cnt  = sums + (size_t)NUM_VOX * 4;   // [NUM_VOX]

    // zero accumulators + canvas (canvas zero == empty-pillar output, and is
    // the identity for the u32 atomic-max over non-negative floats)
    hipMemsetAsync(d_ws, 0, (size_t)NUM_VOX * 5 * sizeof(float), stream);
    hipMemsetAsync(d_out, 0, (size_t)out_size * sizeof(float), stream);

    pillar_scatter_add<<<2048, 256, 0, stream>>>(feat, coors, sums, cnt, n);

    // 768 blocks * 8 waves = 6144 waves grid-striding over 125k point-groups
    pfn_wmma_fused<<<768, 256, 0, stream>>>(feat, coors, W, gamma, beta,
                                            bmean, bvar, sums, cnt, out, n);
}